// GATLayer_35562329210947
// MI455X (gfx1250) — compile-verified
//
#include <hip/hip_runtime.h>

typedef __attribute__((ext_vector_type(2))) float v2f;
typedef __attribute__((ext_vector_type(8))) float v8f;

#define KDIM   256   // IN_FEATS
#define HF     128   // NUM_HEADS * OUT_FEATS
#define NHEADS 4
#define FHEAD  32
#define ASTRIDE 260  // padded LDS row stride (dwords); 260 % 64 == 4 -> conflict-free

// Order-preserving float->uint encoding for atomic max on floats.
__device__ __forceinline__ unsigned fenc(float x) {
  unsigned u = __float_as_uint(x);
  return (u & 0x80000000u) ? ~u : (u | 0x80000000u);
}
__device__ __forceinline__ float fdec(unsigned u) {
  return (u & 0x80000000u) ? __uint_as_float(u & 0x7FFFFFFFu)
                           : __uint_as_float(~u);
}

__global__ void zero_f32(float* __restrict__ p, long n) {
  long i = (long)blockIdx.x * blockDim.x + threadIdx.x;
  if (i < n) p[i] = 0.f;
}

// FT = Hm @ W : (nrows x 256) x (256 x 128), fp32 via V_WMMA_F32_16X16X4_F32.
// Block = 8 waves; the 16x256 A tile is staged once in LDS and shared by all
// waves; wave w computes the 16x16 tile for columns [16w, 16w+16).
__global__ __launch_bounds__(256) void gemm_proj(const float* __restrict__ Hm,
                                                 const float* __restrict__ W,
                                                 float* __restrict__ FT,
                                                 int nrows) {
  __shared__ float As[16 * ASTRIDE];

  const int tid   = threadIdx.x;
  const int mbase = blockIdx.x * 16;

  // Cooperative A-tile load: 16 rows x 256 cols; each thread moves 16 floats.
  {
    int r  = tid >> 4;           // 0..15
    int c0 = (tid & 15) * 16;    // 16 contiguous floats
    int row = mbase + r;
    if (row >= nrows) row = nrows - 1;  // clamp (dup rows never stored)
    const float* src = Hm + (size_t)row * KDIM + c0;
    float4 v0 = *(const float4*)(src + 0);
    float4 v1 = *(const float4*)(src + 4);
    float4 v2 = *(const float4*)(src + 8);
    float4 v3 = *(const float4*)(src + 12);
    float* dst = &As[r * ASTRIDE + c0];
    *(float4*)(dst + 0)  = v0;
    *(float4*)(dst + 4)  = v1;
    *(float4*)(dst + 8)  = v2;
    *(float4*)(dst + 12) = v3;
  }
  __syncthreads();

  const int wave  = tid >> 5;
  const int lane  = tid & 31;
  const int half  = lane >> 4;   // 0: K={k,k+1}   1: K={k+2,k+3}
  const int l16   = lane & 15;
  const int ncol  = wave * 16 + l16;   // 8 waves -> 128 cols
  const int khalf = half * 2;
  const float* arow = &As[l16 * ASTRIDE];

  v8f acc = {};
#pragma unroll 4
  for (int k = 0; k < KDIM; k += 4) {
    v2f a, b;
    // A 16x4 fragment from LDS (bank-conflict-free via padded stride)
    a.x = arow[k + khalf];
    a.y = arow[k + khalf + 1];
    // B 4x16 fragment from global (W is L2-resident, coalesced per 16 lanes)
    b.x = W[(size_t)(k + khalf)     * HF + ncol];
    b.y = W[(size_t)(k + khalf + 1) * HF + ncol];
    acc = __builtin_amdgcn_wmma_f32_16x16x4_f32(false, a, false, b,
                                                (short)0, acc, false, false);
  }

  // C/D layout: VGPR i -> M = i (lanes0-15) / i+8 (lanes16-31), N = l16
  const int rbase = mbase + half * 8;
  float* outp = FT + (size_t)rbase * HF + ncol;
  if (mbase + 16 <= nrows) {      // full tile: branchless stores
#pragma unroll
    for (int i = 0; i < 8; ++i) outp[(size_t)i * HF] = acc[i];
  } else {                        // ragged last tile
#pragma unroll
    for (int i = 0; i < 8; ++i)
      if (rbase + i < nrows) outp[(size_t)i * HF] = acc[i];
  }
}

// out[n,h] = dot(FT[n, h*32 .. h*32+31], attn[h, :])
__global__ void node_logits(const float* __restrict__ FT,
                            const float* __restrict__ attn,
                            float* __restrict__ out, int n) {
  int idx = blockIdx.x * blockDim.x + threadIdx.x;
  if (idx >= n * NHEADS) return;
  int node = idx >> 2, h = idx & 3;
  const float* f = FT + (size_t)node * HF + h * FHEAD;
  const float* a = attn + h * FHEAD;
  float s = 0.f;
#pragma unroll
  for (int i = 0; i < FHEAD; i += 4) {
    float4 fv = *(const float4*)(f + i);
    float4 av = *(const float4*)(a + i);
    s += fv.x * av.x + fv.y * av.y + fv.z * av.z + fv.w * av.w;
  }
  out[idx] = s;
}

// e = leaky_relu(el[src] + er[dst]); segment-max into smax (encoded uint)
__global__ void edge_max(const float* __restrict__ el, const float* __restrict__ er,
                         const int* __restrict__ src, const int* __restrict__ dst,
                         float* __restrict__ e, unsigned* __restrict__ smax, int E) {
  int i = blockIdx.x * blockDim.x + threadIdx.x;
  if (i >= E) return;
  int s = src[i], d = dst[i];
  float4 l = *(const float4*)(el + (size_t)s * NHEADS);
  float4 r = *(const float4*)(er + (size_t)d * NHEADS);
  float v[4] = {l.x + r.x, l.y + r.y, l.z + r.z, l.w + r.w};
#pragma unroll
  for (int h = 0; h < NHEADS; ++h) {
    float x = v[h] > 0.f ? v[h] : 0.2f * v[h];
    e[(size_t)i * NHEADS + h] = x;
    atomicMax(&smax[(size_t)d * NHEADS + h], fenc(x));
  }
}

// p = exp(e - smax[dst]); segment-sum into ssum; overwrite e with p
__global__ void edge_exp(const int* __restrict__ dst, float* __restrict__ e,
                         const unsigned* __restrict__ smax,
                         float* __restrict__ ssum, int E) {
  int i = blockIdx.x * blockDim.x + threadIdx.x;
  if (i >= E) return;
  int d = dst[i];
  float4 ev = *(const float4*)(e + (size_t)i * NHEADS);
  uint4  mv = *(const uint4*)(smax + (size_t)d * NHEADS);
  float p0 = __expf(ev.x - fdec(mv.x));
  float p1 = __expf(ev.y - fdec(mv.y));
  float p2 = __expf(ev.z - fdec(mv.z));
  float p3 = __expf(ev.w - fdec(mv.w));
  *(float4*)(e + (size_t)i * NHEADS) = make_float4(p0, p1, p2, p3);
  atomicAdd(&ssum[(size_t)d * NHEADS + 0], p0);
  atomicAdd(&ssum[(size_t)d * NHEADS + 1], p1);
  atomicAdd(&ssum[(size_t)d * NHEADS + 2], p2);
  atomicAdd(&ssum[(size_t)d * NHEADS + 3], p3);
}

// out[dst] += ft_src[src] * (p / ssum[dst]); one wave (32 lanes x float4) per edge
__global__ void edge_scatter(const float* __restrict__ ft_src,
                             const float* __restrict__ p,
                             const float* __restrict__ ssum,
                             const int* __restrict__ src, const int* __restrict__ dst,
                             float* __restrict__ out, int E) {
  long tid = (long)blockIdx.x * blockDim.x + threadIdx.x;
  int i = (int)(tid >> 5);        // edge id (lanes of a wave share one edge)
  int g = (int)(tid & 31);        // 32 groups of 4 columns = 128 cols
  if (i >= E) return;
  int s = src[i], d = dst[i];
  int col = g * 4;
  int h = col >> 5;               // head for this 4-column group
  float a = p[(size_t)i * NHEADS + h] / ssum[(size_t)d * NHEADS + h];
  float4 f = *(const float4*)(ft_src + (size_t)s * HF + col);
  float* o = out + (size_t)d * HF + col;
  atomicAdd(o + 0, f.x * a);
  atomicAdd(o + 1, f.y * a);
  atomicAdd(o + 2, f.z * a);
  atomicAdd(o + 3, f.w * a);
}

extern "C" void kernel_launch(void* const* d_in, const int* in_sizes, int n_in,
                              void* d_out, int out_size, void* d_ws, size_t ws_size,
                              hipStream_t stream) {
  const float* h_src  = (const float*)d_in[0];
  const float* h_dst  = (const float*)d_in[1];
  const float* W      = (const float*)d_in[2];
  const float* attn_l = (const float*)d_in[3];
  const float* attn_r = (const float*)d_in[4];
  const int*   src_i  = (const int*)d_in[5];
  const int*   dst_i  = (const int*)d_in[6];

  const int n_src = in_sizes[0] / KDIM;
  const int n_dst = in_sizes[1] / KDIM;
  const int E     = in_sizes[5];

  float* ws = (float*)d_ws;
  float* ft_src = ws;              ws += (size_t)n_src * HF;
  float* ft_dst = ws;              ws += (size_t)n_dst * HF;
  float* el     = ws;              ws += (size_t)n_src * NHEADS;
  float* er     = ws;              ws += (size_t)n_dst * NHEADS;
  float* ebuf   = ws;              ws += (size_t)E * NHEADS;
  unsigned* smax = (unsigned*)ws;  ws += (size_t)n_dst * NHEADS;
  float* ssum   = ws;              // contiguous after smax

  float* out = (float*)d_out;

  long zn_out = (long)n_dst * HF;
  zero_f32<<<(int)((zn_out + 255) / 256), 256, 0, stream>>>(out, zn_out);
  long zn_stats = (long)n_dst * NHEADS * 2;  // smax + ssum (contiguous)
  zero_f32<<<(int)((zn_stats + 255) / 256), 256, 0, stream>>>((float*)smax, zn_stats);

  gemm_proj<<<(n_src + 15) / 16, 256, 0, stream>>>(h_src, W, ft_src, n_src);
  gemm_proj<<<(n_dst + 15) / 16, 256, 0, stream>>>(h_dst, W, ft_dst, n_dst);

  node_logits<<<(n_src * NHEADS + 255) / 256, 256, 0, stream>>>(ft_src, attn_l, el, n_src);
  node_logits<<<(n_dst * NHEADS + 255) / 256, 256, 0, stream>>>(ft_dst, attn_r, er, n_dst);

  edge_max<<<(E + 255) / 256, 256, 0, stream>>>(el, er, src_i, dst_i, ebuf, smax, E);
  edge_exp<<<(E + 255) / 256, 256, 0, stream>>>(dst_i, ebuf, smax, ssum, E);

  long sn = (long)E * 32;
  edge_scatter<<<(int)((sn + 255) / 256), 256, 0, stream>>>(ft_src, ebuf, ssum,
                                                            src_i, dst_i, out, E);
}